// PointPillarsScatter_214748364887
// MI455X (gfx1250) — compile-verified
//
#include <hip/hip_runtime.h>

// Problem constants (match reference; static for this dispatch)
#define NUM_FEATURES 64
#define CANVAS_H     496
#define CANVAS_W     432
#define HW           (CANVAS_H * CANVAS_W)       // 214272
#define PLANE_B      (NUM_FEATURES * HW)         // 13713408 floats per batch
#define NUM_PILLARS  48000                       // divisible by 4

typedef __attribute__((ext_vector_type(4))) float v4f;

// ---------------------------------------------------------------------------
// Scatter-add: 256 threads/block = 8 wave32s = 4 pillars x 64 features.
//   thread t -> pillar p = t>>6, feature f = t&63
//   feats flat index == global thread id -> fully coalesced b32 loads.
// 4 lanes decode the 4 pillar rows (16B each) and broadcast the base offset
// through LDS. One hardware global_atomic_add_f32 per (pillar, feature);
// atomics are required because duplicate (b,y,x) cells exist.
// NUM_PILLARS % 4 == 0, grid sized exactly -> no bounds guards needed.
// ---------------------------------------------------------------------------
__global__ void pp_scatter_add(const float* __restrict__ feats,
                               const int*   __restrict__ coors,
                               float*       __restrict__ out) {
    __shared__ int sbase[4];

    const int tid = threadIdx.x;
    const int p0  = blockIdx.x * 4;

    if (tid < 4) {
        // coors row = [batch, z, y, x], 16B aligned
        const int4 c = ((const int4*)coors)[p0 + tid];
        sbase[tid] = c.x * PLANE_B + c.z * CANVAS_W + c.w;  // fits in int32
    }
    __syncthreads();

    // feats[(p0+lp)*64 + f] == feats[blockIdx.x*256 + tid] -> coalesced
    const float v = feats[blockIdx.x * 256 + tid];

    // out[b, f, y, x] += v  (hardware fp32 atomic, no-return -> STOREcnt)
    unsafeAtomicAdd(out + sbase[tid >> 6] + (tid & 63) * HW, v);
}

// ---------------------------------------------------------------------------
// Zero the 219 MB canvas with 16B stores (global_store_b128).
// Default (RT / wr-rinse) temporal hint keeps zeroed lines valid in the
// 192 MB L2 so the scatter kernel's atomics mostly hit in L2.
// ---------------------------------------------------------------------------
__global__ void pp_zero_canvas(float* __restrict__ out, int n4) {
    int i      = blockIdx.x * blockDim.x + threadIdx.x;
    int stride = gridDim.x * blockDim.x;
    v4f z = (v4f)0.0f;
    v4f* __restrict__ o4 = (v4f*)out;
    for (; i < n4; i += stride) {
        o4[i] = z;
    }
}

// ---------------------------------------------------------------------------
// Harness entry point.
//   d_in[0] : pillar_features, float32, 48000*64
//   d_in[1] : coors,           int32,   48000*4
//   d_out   : float32, 4*64*496*432 = 54,853,632
// Zero-fill MUST run every call: atomics accumulate and the harness replays
// the captured graph without re-poisoning d_out.
// ---------------------------------------------------------------------------
extern "C" void kernel_launch(void* const* d_in, const int* in_sizes, int n_in,
                              void* d_out, int out_size, void* d_ws, size_t ws_size,
                              hipStream_t stream) {
    const float* feats = (const float*)d_in[0];
    const int*   coors = (const int*)  d_in[1];
    float*       out   = (float*)d_out;

    // ---- zero canvas: out_size floats, 4 float4 stores per thread ----
    const int n4       = out_size >> 2;                      // float4 count
    const int zthreads = 256;
    int zblocks = (n4 + zthreads * 4 - 1) / (zthreads * 4);  // ~13392
    if (zblocks < 1) zblocks = 1;
    pp_zero_canvas<<<zblocks, zthreads, 0, stream>>>(out, n4);

    // ---- scatter: 4 pillars per 256-thread block (48000/4 = 12000 blocks) --
    const int n_pillars = in_sizes[1] / 4;                   // = 48000
    const int sblocks   = n_pillars / 4;
    pp_scatter_add<<<sblocks, 256, 0, stream>>>(feats, coors, out);
}